// VQVAE_52733608460736
// MI455X (gfx1250) — compile-verified
//
#include <hip/hip_runtime.h>

// Problem dims
#define BB 8
#define CC 1024
#define LL 2048
#define DD 512
#define KK 4096
#define BL (BB * LL)        // 16384 rows (b,l)

// Search kernel A-tile: d-major [512][80]; pitch 80 -> async b128 staging is a
// contiguous copy, and frag reads are conflict-free (2*80 % 64 == 32).
#define SPS 80
#define LDS_SEARCH (DD * SPS * 4)        // 160 KB
// Decoder Q-tile: l-major [64][516]; pitch 516 -> conflict-free b64 frag reads.
#define SPQ 516
#define LDS_DEC (64 * SPQ * 4)           // ~129 KB

typedef __attribute__((ext_vector_type(2))) float vf2;
typedef __attribute__((ext_vector_type(8))) float v8f;

// D = A(16x4 f32) * B(4x16 f32) + C(16x16 f32), full fp32 matrix op
__device__ __forceinline__ v8f wmma4(vf2 a, vf2 b, v8f c) {
    return __builtin_amdgcn_wmma_f32_16x16x4_f32(
        /*neg_a=*/false, a, /*neg_b=*/false, b,
        /*c_mod=*/(short)0, c, /*reuse_a=*/false, /*reuse_b=*/false);
}

// Async copy: 16 bytes per lane, global -> LDS (per-lane LDS byte offset).
__device__ __forceinline__ void async_copy_b128(unsigned lds_byte_off, const float* gptr) {
    asm volatile("global_load_async_to_lds_b128 %0, %1, off"
                 :: "v"(lds_byte_off), "v"(gptr) : "memory");
}
__device__ __forceinline__ void async_wait0() {
    asm volatile("s_wait_asynccnt 0x0" ::: "memory");
}

// ---------------------------------------------------------------- init
__global__ void k_init(float* acc) {
    if (threadIdx.x == 0) { acc[0] = 0.f; acc[1] = 0.f; }
}

// ---------------------------------------------------------------- transpose src[R][S] -> dst[S][R]
__global__ void k_transpose(const float* __restrict__ src, float* __restrict__ dst,
                            int R, int S) {
    __shared__ float tile[32][33];
    int c = blockIdx.x * 32 + threadIdx.x;
    int r0 = blockIdx.y * 32;
    for (int j = threadIdx.y; j < 32; j += 8) {
        int r = r0 + j;
        if (r < R && c < S) tile[j][threadIdx.x] = src[(size_t)r * S + c];
    }
    __syncthreads();
    int rr = r0 + threadIdx.x;
    int s0 = blockIdx.x * 32;
    for (int j = threadIdx.y; j < 32; j += 8) {
        int s = s0 + j;
        if (s < S && rr < R) dst[(size_t)s * R + rr] = tile[threadIdx.x][j];
    }
}

// ---------------------------------------------------------------- e_sq[k] = ||codebook[k]||^2  (one wave per k)
__global__ void k_esq(const float* __restrict__ cb, float* __restrict__ esq) {
    int lane = threadIdx.x & 31;
    int k = blockIdx.x * 4 + (threadIdx.x >> 5);
    float s = 0.f;
    for (int j = 0; j < DD; j += 32) {
        float v = cb[(size_t)k * DD + j + lane];
        s += v * v;
    }
    #pragma unroll
    for (int m = 16; m >= 1; m >>= 1) s += __shfl_xor(s, m, 32);
    if (lane == 0) esq[k] = s;
}

// ---------------------------------------------------------------- encoder: zT[d][bl] = enc_w . x + enc_b
// One wave per 16(d) x 16(bl) tile, K-loop over C with fp32 WMMA.
__global__ void k_encoder(const float* __restrict__ x, const float* __restrict__ encwT,
                          const float* __restrict__ encb, float* __restrict__ zT) {
    int wid = (blockIdx.x * blockDim.x + threadIdx.x) >> 5;
    int lane = threadIdx.x & 31;
    int half = lane >> 4;
    int l15 = lane & 15;
    int d0  = (wid & 31) * 16;            // 32 d-tiles
    int bl0 = (wid >> 5) * 16;            // 1024 bl-tiles
    int b  = bl0 >> 11;                   // bl0 / L
    int l0 = bl0 & (LL - 1);
    const float* xb = x + (size_t)b * CC * LL + l0;
    const int ka = 2 * half;

    v8f acc = {0, 0, 0, 0, 0, 0, 0, 0};
    for (int c0 = 0; c0 < CC; c0 += 4) {
        vf2 a, bf;
        a.x = encwT[(size_t)(c0 + ka) * DD + d0 + l15];
        a.y = encwT[(size_t)(c0 + ka + 1) * DD + d0 + l15];
        bf.x = xb[(size_t)(c0 + ka) * LL + l15];
        bf.y = xb[(size_t)(c0 + ka + 1) * LL + l15];
        acc = wmma4(a, bf, acc);
    }
    #pragma unroll
    for (int r = 0; r < 8; ++r) {
        int dr = d0 + r + 8 * half;
        zT[(size_t)dr * BL + bl0 + l15] = acc[r] + encb[dr];
    }
}

// ---------------------------------------------------------------- sum of |z|^2 into loss accumulator
__global__ void k_zsq(const float* __restrict__ zT, float* __restrict__ acc) {
    int bl = blockIdx.x * blockDim.x + threadIdx.x;
    float s = 0.f;
    for (int d = 0; d < DD; ++d) {
        float v = zT[(size_t)d * BL + bl];
        s += v * v;
    }
    #pragma unroll
    for (int m = 16; m >= 1; m >>= 1) s += __shfl_xor(s, m, 32);
    if ((threadIdx.x & 31) == 0) atomicAdd(acc, s);
}

// ---------------------------------------------------------------- codebook search (argmin) + loss partial
// Block = 4 waves, 64 bl rows. z-tile async-copied into LDS (d-major, pitch SPS);
// codebookT streamed (L2-resident).
__global__ void k_search(const float* __restrict__ zT, const float* __restrict__ cbT,
                         const float* __restrict__ esq, int* __restrict__ idxw,
                         float* __restrict__ out_idx, float* __restrict__ acc) {
    extern __shared__ float Asl[];        // [DD][SPS]
    const int bl0 = blockIdx.x * 64;
    const int tid = threadIdx.x;

    // Stage zT tile with async global->LDS b128: contiguous 16B per lane.
    for (int u = tid * 4; u < 64 * DD; u += 128 * 4) {
        int d = u >> 6;                   // 64 bl per d-row
        int m = u & 63;
        async_copy_b128((unsigned)((d * SPS + m) * 4),
                        zT + (size_t)d * BL + bl0 + m);
    }
    async_wait0();
    __syncthreads();

    const int lane = tid & 31;
    const int half = lane >> 4;
    const int l15 = lane & 15;
    const int rb = (tid >> 5) * 16;       // wave's 16-row slice
    const int ka = 2 * half;

    float bv[8];
    int bi[8];
    #pragma unroll
    for (int r = 0; r < 8; ++r) { bv[r] = 3.0e38f; bi[r] = 0; }

    const float* Arow = Asl + rb + l15;   // + d*SPS indexes the d'th row

    for (int k0 = 0; k0 < KK; k0 += 16) {
        float es = esq[k0 + l15];
        v8f dots = {0, 0, 0, 0, 0, 0, 0, 0};
        const float* Bp = cbT + (size_t)ka * KK + k0 + l15;
        #pragma unroll 4
        for (int d0 = 0; d0 < DD; d0 += 4) {
            if ((d0 & 63) == 0)
                __builtin_prefetch(Bp + (size_t)(d0 + 64) * KK, 0, 0);
            vf2 a;
            a.x = Arow[(d0 + ka) * SPS];      // ds_load_b32, conflict-free
            a.y = Arow[(d0 + ka + 1) * SPS];
            vf2 bf;
            bf.x = Bp[(size_t)d0 * KK];
            bf.y = Bp[(size_t)(d0 + 1) * KK];
            dots = wmma4(a, bf, dots);
        }
        #pragma unroll
        for (int r = 0; r < 8; ++r) {
            float sc = es - 2.0f * dots[r];
            if (sc < bv[r]) { bv[r] = sc; bi[r] = k0 + l15; }
        }
    }

    // min-reduce (value, index) across the 16 lane-columns inside each half-wave
    #pragma unroll
    for (int m = 1; m <= 8; m <<= 1) {
        #pragma unroll
        for (int r = 0; r < 8; ++r) {
            float ov = __shfl_xor(bv[r], m, 32);
            int oi = __shfl_xor(bi[r], m, 32);
            if (ov < bv[r] || (ov == bv[r] && oi < bi[r])) { bv[r] = ov; bi[r] = oi; }
        }
    }
    if (l15 == 0) {
        float ls = 0.f;
        #pragma unroll
        for (int r = 0; r < 8; ++r) {
            int row = bl0 + rb + r + 8 * half;
            idxw[row] = bi[r];
            out_idx[row] = (float)bi[r];
            ls += bv[r];                  // min(e_sq - 2*dots); |z|^2 added by k_zsq
        }
        atomicAdd(acc, ls);
    }
}

// ---------------------------------------------------------------- decoder: recon[b][c][l] = dec_w . codebook[idx] + dec_b
// Block = 4 waves, 64(c) x 64(l) output tile; gathered codebook rows async-copied to LDS.
__global__ void k_decoder(const float* __restrict__ cb, const int* __restrict__ idxw,
                          const float* __restrict__ decwT, const float* __restrict__ decb,
                          float* __restrict__ out) {
    extern __shared__ float Q[];          // [64 l-rows][SPQ]
    const int bl0 = blockIdx.x * 64;
    const int c0 = blockIdx.y * 64;
    const int b = bl0 >> 11;
    const int l0 = bl0 & (LL - 1);
    const int tid = threadIdx.x;

    // Gather quantized rows: each pass the 128 threads stream one full codebook row.
    for (int u = tid * 4; u < 64 * DD; u += 128 * 4) {
        int l = u >> 9;
        int d = u & (DD - 1);
        int ci = idxw[bl0 + l];
        async_copy_b128((unsigned)((l * SPQ + d) * 4),
                        cb + (size_t)ci * DD + d);
    }
    async_wait0();
    __syncthreads();

    const int lane = tid & 31;
    const int half = lane >> 4;
    const int l15 = lane & 15;
    const int w = tid >> 5;               // wave id -> c sub-tile
    const int cw = c0 + 16 * w;
    const int ka = 2 * half;

    v8f acc0 = {0,0,0,0,0,0,0,0}, acc1 = {0,0,0,0,0,0,0,0};
    v8f acc2 = {0,0,0,0,0,0,0,0}, acc3 = {0,0,0,0,0,0,0,0};

    for (int d0 = 0; d0 < DD; d0 += 4) {
        vf2 a;
        a.x = decwT[(size_t)(d0 + ka) * CC + cw + l15];
        a.y = decwT[(size_t)(d0 + ka + 1) * CC + cw + l15];
        vf2 b0 = *reinterpret_cast<const vf2*>(&Q[(0 * 16 + l15) * SPQ + d0 + ka]);
        vf2 b1 = *reinterpret_cast<const vf2*>(&Q[(1 * 16 + l15) * SPQ + d0 + ka]);
        vf2 b2 = *reinterpret_cast<const vf2*>(&Q[(2 * 16 + l15) * SPQ + d0 + ka]);
        vf2 b3 = *reinterpret_cast<const vf2*>(&Q[(3 * 16 + l15) * SPQ + d0 + ka]);
        acc0 = wmma4(a, b0, acc0);        // A reused across 4 l sub-tiles
        acc1 = wmma4(a, b1, acc1);
        acc2 = wmma4(a, b2, acc2);
        acc3 = wmma4(a, b3, acc3);
    }

    float* ob = out + (size_t)b * CC * LL;
    #pragma unroll
    for (int r = 0; r < 8; ++r) {
        int c = cw + r + 8 * half;
        float bias = decb[c];
        ob[(size_t)c * LL + l0 + 0 * 16 + l15] = acc0[r] + bias;
        ob[(size_t)c * LL + l0 + 1 * 16 + l15] = acc1[r] + bias;
        ob[(size_t)c * LL + l0 + 2 * 16 + l15] = acc2[r] + bias;
        ob[(size_t)c * LL + l0 + 3 * 16 + l15] = acc3[r] + bias;
    }
}

// ---------------------------------------------------------------- finalize loss
__global__ void k_final(const float* __restrict__ acc, float* __restrict__ out_loss) {
    if (threadIdx.x == 0) out_loss[0] = acc[0] / (float)(BL * DD);
}

// ---------------------------------------------------------------- host
extern "C" void kernel_launch(void* const* d_in, const int* in_sizes, int n_in,
                              void* d_out, int out_size, void* d_ws, size_t ws_size,
                              hipStream_t stream) {
    const float* x    = (const float*)d_in[0];   // [B,C,L]
    const float* encw = (const float*)d_in[1];   // [D,C]
    const float* encb = (const float*)d_in[2];   // [D]
    const float* cb   = (const float*)d_in[3];   // [K,D]
    const float* decw = (const float*)d_in[4];   // [C,D]
    const float* decb = (const float*)d_in[5];   // [C]
    float* out = (float*)d_out;                  // recon | indices(float) | loss

    float* ws = (float*)d_ws;
    float* encwT = ws;                           // C*D
    float* decwT = encwT + (size_t)CC * DD;      // D*C
    float* cbT   = decwT + (size_t)DD * CC;      // D*K
    float* esq   = cbT + (size_t)DD * KK;        // K
    float* acc   = esq + KK;                     // 16 (loss accumulator)
    int*   idxw  = (int*)(acc + 16);             // BL
    float* zT    = (float*)(idxw + BL);          // D*BL

    (void)hipFuncSetAttribute((const void*)k_search,
                              hipFuncAttributeMaxDynamicSharedMemorySize, LDS_SEARCH);
    (void)hipFuncSetAttribute((const void*)k_decoder,
                              hipFuncAttributeMaxDynamicSharedMemorySize, LDS_DEC);

    k_init<<<1, 32, 0, stream>>>(acc);

    dim3 tb(32, 8);
    k_transpose<<<dim3(CC / 32, DD / 32), tb, 0, stream>>>(encw, encwT, DD, CC);
    k_transpose<<<dim3(DD / 32, KK / 32), tb, 0, stream>>>(cb, cbT, KK, DD);
    k_transpose<<<dim3(DD / 32, CC / 32), tb, 0, stream>>>(decw, decwT, CC, DD);
    k_esq<<<KK / 4, 128, 0, stream>>>(cb, esq);

    k_encoder<<<(32 * (BL / 16)) / 8, 256, 0, stream>>>(x, encwT, encb, zT);
    k_zsq<<<BL / 256, 256, 0, stream>>>(zT, acc);

    float* out_idx = out + (size_t)BB * CC * LL;
    k_search<<<BL / 64, 128, LDS_SEARCH, stream>>>(zT, cbT, esq, idxw, out_idx, acc);
    k_decoder<<<dim3(BL / 64, CC / 64), 128, LDS_DEC, stream>>>(cb, idxw, decwT, decb, out);
    k_final<<<1, 32, 0, stream>>>(acc, out + (size_t)BB * CC * LL + BL);
}